// CosyVoice2Attention_91225105367697
// MI455X (gfx1250) — compile-verified
//
#include <hip/hip_runtime.h>
#include <math.h>

typedef __attribute__((ext_vector_type(2))) float v2f;
typedef __attribute__((ext_vector_type(8))) float v8f;
typedef __attribute__((ext_vector_type(4))) unsigned int u32x4;
typedef __attribute__((ext_vector_type(4))) int i32x4;
typedef __attribute__((ext_vector_type(8))) int i32x8;

#define HIDDEN    896
#define NQH       14
#define NKVH      2
#define HD        64
#define BB        8
#define SS        2048
#define MTOT      (BB*SS)        // 16384 tokens
#define KVW       (NKVH*HD)      // 128
#define GROUP     (NQH/NKVH)     // 7
#define ROPE_HALF 32

union F8 { v8f v; float f[8]; };

// reductions within each 16-lane half of a wave32 (matches WMMA C-layout rows)
__device__ __forceinline__ float half16_max(float v) {
  v = fmaxf(v, __shfl_xor(v, 8, 32));
  v = fmaxf(v, __shfl_xor(v, 4, 32));
  v = fmaxf(v, __shfl_xor(v, 2, 32));
  v = fmaxf(v, __shfl_xor(v, 1, 32));
  return v;
}
__device__ __forceinline__ float half16_sum(float v) {
  v += __shfl_xor(v, 8, 32);
  v += __shfl_xor(v, 4, 32);
  v += __shfl_xor(v, 2, 32);
  v += __shfl_xor(v, 1, 32);
  return v;
}

// ---------------------------------------------------------------------------
// Tensor Data Mover: 2D f32 tile (tile_d0 cols x tile_d1 rows) from global
// (row stride = stride_d0 elements) into LDS at lds_off, with 1 DWORD of LDS
// padding inserted every 32 DWORDs (pad_interval=4 -> 2^(4+1)=32), so a
// 32-float tile row lands with LDS stride 33 (conflict-free WMMA frag reads).
// D# layout per CDNA5 ISA ch.8 (group0 128b, group1 256b; groups 2/3 zero).
// ---------------------------------------------------------------------------
__device__ __forceinline__ void tdm_load_2d(unsigned lds_off, const float* gptr,
                                            unsigned tensor_d0, unsigned tensor_d1,
                                            unsigned tile_d0, unsigned tile_d1,
                                            unsigned stride_d0)
{
  const unsigned long long ga = (unsigned long long)(size_t)gptr;
  u32x4 g0;
  g0.x = 1u;                                           // count=1 valid, user desc
  g0.y = lds_off;                                      // lds_addr[31:0]
  g0.z = (unsigned)(ga & 0xFFFFFFFFu);                 // global_addr[31:0]
  g0.w = (unsigned)((ga >> 32) & 0x1FFFFFFu) | (2u << 30); // addr[56:32] | type=2
  i32x8 g1;
  g1[0] = (int)((2u << 16) | (1u << 20) | (4u << 22)); // data_size=4B, pad 1dw/32dw
  g1[1] = (int)((tensor_d0 & 0xFFFFu) << 16);          // tensor_dim0[15:0]
  g1[2] = (int)((tensor_d0 >> 16) | ((tensor_d1 & 0xFFFFu) << 16));
  g1[3] = (int)((tensor_d1 >> 16) | (tile_d0 << 16));  // tile_dim0
  g1[4] = (int)tile_d1;                                // tile_dim1 (tile_dim2=0)
  g1[5] = (int)stride_d0;                              // tensor_dim0_stride[31:0]
  g1[6] = 0;                                           // stride hi / dim1_stride lo
  g1[7] = 0;
  i32x4 z4 = {0, 0, 0, 0};
#if defined(__clang_major__) && (__clang_major__ >= 23)
  i32x8 z8 = {0, 0, 0, 0, 0, 0, 0, 0};
  __builtin_amdgcn_tensor_load_to_lds(g0, g1, z4, z4, z8, 0);
#else
  __builtin_amdgcn_tensor_load_to_lds(g0, g1, z4, z4, 0);
#endif
}

// ---------------------------------------------------------------------------
// C[M,N] = X[M,K] @ W[K,N] (+ bias).  Block tile 64x32, 256 threads = 8 waves,
// each wave owns one 16x16 C tile (V_WMMA_F32_16X16X4_F32).
// Tiles are staged via TENSOR_LOAD_TO_LDS (TDM), double-buffered on TENSORcnt.
// M = MTOT, K % 32 == 0, N % 32 == 0.
// ---------------------------------------------------------------------------
__global__ void __launch_bounds__(256)
gemm_f32_wmma(const float* __restrict__ X, const float* __restrict__ W,
              const float* __restrict__ bias, float* __restrict__ C,
              int K, int N)
{
  __shared__ float As[2][64 * 33];          // [m][k] rows padded to 33 by TDM
  __shared__ float Bs[2][32 * 33];          // [k][n] rows padded to 33 by TDM

  const int t    = threadIdx.x;
  const int lane = t & 31;
  const int wave = t >> 5;
  const int wm   = wave & 3;                // 4 wave rows (m)
  const int wn   = wave >> 2;               // 2 wave cols (n)
  const int lh   = lane >> 4;               // half-wave id (K split)
  const int ll   = lane & 15;               // row/col within fragment
  const int m0   = blockIdx.y * 64;
  const int n0   = blockIdx.x * 32;

  const unsigned ldsA[2] = { (unsigned)(size_t)&As[0][0], (unsigned)(size_t)&As[1][0] };
  const unsigned ldsB[2] = { (unsigned)(size_t)&Bs[0][0], (unsigned)(size_t)&Bs[1][0] };

  auto issue_chunk = [&](int buf, int k0) {
    // A tile: 32(k) x 64(m) from X row-major (row stride K)
    tdm_load_2d(ldsA[buf], X + (size_t)m0 * K + k0,
                (unsigned)K, (unsigned)MTOT, 32u, 64u, (unsigned)K);
    // B tile: 32(n) x 32(k) from W row-major (row stride N)
    tdm_load_2d(ldsB[buf], W + (size_t)k0 * N + n0,
                (unsigned)N, (unsigned)K, 32u, 32u, (unsigned)N);
  };

  const int nchunks = K / 32;
  if (t < 32) issue_chunk(0, 0);            // wave 0 drives the TDM pipeline

  F8 acc = {};

  for (int i = 0; i < nchunks; ++i) {
    if (t < 32) {
      if (i + 1 < nchunks) {
        issue_chunk((i + 1) & 1, (i + 1) * 32);       // prefetch next chunk
        __builtin_amdgcn_s_wait_tensorcnt((short)2);  // chunk i complete (in-order)
      } else {
        __builtin_amdgcn_s_wait_tensorcnt((short)0);
      }
    }
    __syncthreads();

    const float* Asb = &As[i & 1][0];
    const float* Bsb = &Bs[i & 1][0];
#pragma unroll
    for (int kk = 0; kk < 32; kk += 4) {
      const int ka = kk + lh * 2;
      v2f a, b;
      const float* Ap = Asb + (wm * 16 + ll) * 33;
      a.x = Ap[ka];  a.y = Ap[ka + 1];
      const float* Bp = Bsb + ka * 33 + wn * 16 + ll;
      b.x = Bp[0];   b.y = Bp[33];
      acc.v = __builtin_amdgcn_wmma_f32_16x16x4_f32(
          false, a, false, b, (short)0, acc.v, false, false);
    }
    __syncthreads();                        // tile reads done before next DMA lands
  }

  const int col  = n0 + wn * 16 + ll;
  const int row0 = m0 + wm * 16 + lh * 8;
  const float bv = bias ? bias[col] : 0.0f;
#pragma unroll
  for (int r = 0; r < 8; ++r)
    C[(size_t)(row0 + r) * N + col] = acc.f[r] + bv;
}

// ---------------------------------------------------------------------------
// RoPE on Q (MTOT x 896, 14 heads) and K (MTOT x 128, 2 heads), in place.
// ---------------------------------------------------------------------------
__global__ void __launch_bounds__(256)
rope_kernel(float* __restrict__ Qb, float* __restrict__ Kb,
            const int* __restrict__ pos)
{
  const int idx = blockIdx.x * blockDim.x + threadIdx.x;
  if (idx >= MTOT * (NQH + NKVH) * ROPE_HALF) return;
  const int j   = idx & 31;
  const int h   = (idx >> 5) % (NQH + NKVH);
  const int tok = idx / (32 * (NQH + NKVH));

  const float p  = (float)pos[tok];
  const float iv = powf(1000000.0f, -(float)j * (1.0f / 32.0f));
  const float a  = p * iv;
  const float cs = cosf(a), sn = sinf(a);

  float* base = (h < NQH) ? (Qb + (size_t)tok * HIDDEN + h * HD)
                          : (Kb + (size_t)tok * KVW + (h - NQH) * HD);
  const float x1 = base[j], x2 = base[j + 32];
  base[j]      = x1 * cs - x2 * sn;
  base[j + 32] = x2 * cs + x1 * sn;
}

// ---------------------------------------------------------------------------
// Causal flash attention.  Grid (S/16, NQH, B), 128 threads = 4 waves.
// Each wave: full 16x16 score tile (16 WMMAs over D=64), online softmax via
// half-wave shuffles, then P·V for its private 16-dim output slice (4 WMMAs).
// ---------------------------------------------------------------------------
__global__ void __launch_bounds__(128)
attn_f32_wmma(const float* __restrict__ Q, const float* __restrict__ Kb,
              const float* __restrict__ Vb, float* __restrict__ O)
{
  __shared__ float Qs[16][65];              // [q row][d]
  __shared__ float Ks[16][65];              // [key][d]
  __shared__ __align__(16) float Vs[16][64];// [key][d]
  __shared__ float Ps[16][17];              // softmaxed probs, [q row][key]

  const int qt  = blockIdx.x;
  const int h   = blockIdx.y;
  const int b   = blockIdx.z;
  const int kvh = h / GROUP;
  const int q0  = qt * 16;

  const int t    = threadIdx.x;
  const int lane = t & 31;
  const int wave = t >> 5;                  // 0..3 : 16-dim output slice
  const int lh   = lane >> 4;
  const int ll   = lane & 15;

  // load Q tile
  {
    const int row = t >> 3;                 // 0..15
    const int d   = (t & 7) * 4;            // 0..28
#pragma unroll
    for (int p = 0; p < 2; ++p) {
      const float4 qv = *(const float4*)(
          Q + (size_t)(b * SS + q0 + row) * HIDDEN + h * HD + d + p * 32);
      Qs[row][d + p * 32 + 0] = qv.x; Qs[row][d + p * 32 + 1] = qv.y;
      Qs[row][d + p * 32 + 2] = qv.z; Qs[row][d + p * 32 + 3] = qv.w;
    }
  }

  float row_m[8], row_l[8];
#pragma unroll
  for (int r = 0; r < 8; ++r) { row_m[r] = -__builtin_inff(); row_l[r] = 0.0f; }
  F8 o = {};
  const float sc = 0.125f;                  // 1/sqrt(64)

  for (int kt = 0; kt <= q0; kt += 16) {
    __syncthreads();                        // prev-iter Ks/Vs/Ps reads done
    {
      const int j = t >> 3;
      const int d = (t & 7) * 4;
#pragma unroll
      for (int p = 0; p < 2; ++p) {
        const float4 kv = *(const float4*)(
            Kb + (size_t)(b * SS + kt + j) * KVW + kvh * HD + d + p * 32);
        Ks[j][d + p * 32 + 0] = kv.x; Ks[j][d + p * 32 + 1] = kv.y;
        Ks[j][d + p * 32 + 2] = kv.z; Ks[j][d + p * 32 + 3] = kv.w;
        const float4 vv = *(const float4*)(
            Vb + (size_t)(b * SS + kt + j) * KVW + kvh * HD + d + p * 32);
        *(float4*)&Vs[j][d + p * 32] = vv;
      }
    }
    __syncthreads();

    // S = Q · K^T  (B fragment col j = key index; B[kd][j] = K[j][kd])
    F8 s = {};
#pragma unroll
    for (int dk = 0; dk < HD; dk += 4) {
      const int ka = dk + lh * 2;
      v2f a, bf;
      a.x  = Qs[ll][ka]; a.y  = Qs[ll][ka + 1];
      bf.x = Ks[ll][ka]; bf.y = Ks[ll][ka + 1];
      s.v = __builtin_amdgcn_wmma_f32_16x16x4_f32(
          false, a, false, bf, (short)0, s.v, false, false);
    }

    // online softmax (rows r + 8*lh live in reg r; cols = lanes within half)
    const bool diag = (kt == q0);
#pragma unroll
    for (int r = 0; r < 8; ++r) {
      float sv = s.f[r] * sc;
      const int rowq = r + lh * 8;
      if (diag && (ll > rowq)) sv = -__builtin_inff();
      const float mrow  = half16_max(sv);
      const float mnew  = fmaxf(row_m[r], mrow);
      const float scale = expf(row_m[r] - mnew);
      const float p     = expf(sv - mnew);
      row_l[r] = row_l[r] * scale + half16_sum(p);
      row_m[r] = mnew;
      o.f[r]  *= scale;
      s.f[r]   = p;
    }

    if (wave == 0) {                        // all waves hold identical P
#pragma unroll
      for (int r = 0; r < 8; ++r) Ps[r + lh * 8][ll] = s.f[r];
    }
    __syncthreads();

    // O += P · V  (this wave's 16-dim slice of V)
#pragma unroll
    for (int kk = 0; kk < 16; kk += 4) {
      const int ka = kk + lh * 2;
      v2f a, bf;
      a.x  = Ps[ll][ka];             a.y  = Ps[ll][ka + 1];
      bf.x = Vs[ka][wave * 16 + ll]; bf.y = Vs[ka + 1][wave * 16 + ll];
      o.v = __builtin_amdgcn_wmma_f32_16x16x4_f32(
          false, a, false, bf, (short)0, o.v, false, false);
    }
  }

  // normalize + store attn output in (token, h*64+d) layout for the O-proj GEMM
#pragma unroll
  for (int r = 0; r < 8; ++r) {
    const int rowq = q0 + r + lh * 8;
    O[(size_t)(b * SS + rowq) * HIDDEN + h * HD + wave * 16 + ll] =
        o.f[r] / row_l[r];
  }
}

// ---------------------------------------------------------------------------
extern "C" void kernel_launch(void* const* d_in, const int* in_sizes, int n_in,
                              void* d_out, int out_size, void* d_ws, size_t ws_size,
                              hipStream_t stream)
{
  const float* X   = (const float*)d_in[0];
  const int*   pos = (const int*)  d_in[1];
  const float* Wq  = (const float*)d_in[2];
  const float* bq  = (const float*)d_in[3];
  const float* Wk  = (const float*)d_in[4];
  const float* bk  = (const float*)d_in[5];
  const float* Wv  = (const float*)d_in[6];
  const float* bvp = (const float*)d_in[7];
  const float* Wo  = (const float*)d_in[8];
  float* out = (float*)d_out;

  // workspace: Q(56MB) | K(8MB) | V(8MB) | attn(56MB) = 128MB
  float* Qb = (float*)d_ws;
  float* Kb = Qb + (size_t)MTOT * HIDDEN;
  float* Vb = Kb + (size_t)MTOT * KVW;
  float* Ab = Vb + (size_t)MTOT * KVW;

  gemm_f32_wmma<<<dim3(HIDDEN / 32, MTOT / 64), 256, 0, stream>>>(X, Wq, bq, Qb, HIDDEN, HIDDEN);
  gemm_f32_wmma<<<dim3(KVW    / 32, MTOT / 64), 256, 0, stream>>>(X, Wk, bk, Kb, HIDDEN, KVW);
  gemm_f32_wmma<<<dim3(KVW    / 32, MTOT / 64), 256, 0, stream>>>(X, Wv, bvp, Vb, HIDDEN, KVW);

  const int rope_total = MTOT * (NQH + NKVH) * ROPE_HALF;
  rope_kernel<<<(rope_total + 255) / 256, 256, 0, stream>>>(Qb, Kb, pos);

  attn_f32_wmma<<<dim3(SS / 16, NQH, BB), 128, 0, stream>>>(Qb, Kb, Vb, Ab);

  gemm_f32_wmma<<<dim3(HIDDEN / 32, MTOT / 64), 256, 0, stream>>>(Ab, Wo, nullptr, out, HIDDEN, HIDDEN);
}